// UnsupervisedFlowLosses_29076928594561
// MI455X (gfx1250) — compile-verified
//
#include <hip/hip_runtime.h>

// ---------------- problem constants (fixed by the reference) ----------------
#define N_PTS   16384
#define K_NN    8
#define BLK     256
#define NBLK    (N_PTS / BLK)          // 64 blocks
#define TILE    2048                   // points per LDS tile for KNN scans
#define TILE_F  (TILE * 3)             // 6144 floats = 24 KB
#define VPT_KNN (TILE_F / (4 * BLK))   // 6 float4 per thread per tile
#define TSEG    1024                   // points per tile for segment scan
#define VF_SEG  ((TSEG * 3) / (4 * BLK)) // 3
#define VI_SEG  (TSEG / (4 * BLK))       // 1

// ---------------- CDNA5 feature gates ----------------
#if defined(__gfx1250__) && \
    __has_builtin(__builtin_amdgcn_global_load_async_to_lds_b128) && \
    __has_builtin(__builtin_amdgcn_s_wait_asynccnt)
#define ASYNC_LDS 1
#else
#define ASYNC_LDS 0
#endif

#if ASYNC_LDS
#define ASYNC_WAIT(n) __builtin_amdgcn_s_wait_asynccnt(n)
#else
#define ASYNC_WAIT(n) ((void)0)
#endif

typedef int   v4i __attribute__((vector_size(16)));   // matches builtin's V4i
typedef __attribute__((ext_vector_type(2))) float v2f;
typedef __attribute__((ext_vector_type(8))) float v8f;

// Copy VPT*16 bytes/thread from global to LDS (async on CDNA5, sync fallback).
template <int VPT>
__device__ __forceinline__ void tile_fetch(float* __restrict__ lds,
                                           const float* __restrict__ g,
                                           int tid) {
#if ASYNC_LDS
#pragma unroll
  for (int v = 0; v < VPT; ++v) {
    const float* src = g + (v * BLK + tid) * 4;
    float* dst = lds + (v * BLK + tid) * 4;
    __builtin_amdgcn_global_load_async_to_lds_b128(
        (__attribute__((address_space(1))) v4i*)src,
        (__attribute__((address_space(3))) v4i*)dst, 0, 0);
  }
#else
#pragma unroll
  for (int v = 0; v < VPT; ++v)
    ((float4*)lds)[v * BLK + tid] = ((const float4*)g)[v * BLK + tid];
#endif
}

// Wave-wide (32-lane) sum. Uses the matrix unit as a reduction engine:
// A = ones(16x4), B(VGPR0)=per-lane value, B(VGPR1)=0  =>  D[m,n] = x_n + x_{n+16}.
// One V_WMMA_F32_16X16X4_F32 halves 32->16, four shuffles finish.
__device__ __forceinline__ float wave_sum32(float x) {
#if __has_builtin(__builtin_amdgcn_wmma_f32_16x16x4_f32)
  v2f a; a[0] = 1.0f; a[1] = 1.0f;
  v2f b; b[0] = x;    b[1] = 0.0f;
  v8f c = {};
  c = __builtin_amdgcn_wmma_f32_16x16x4_f32(false, a, false, b,
                                            (short)0, c, false, false);
  float s = c[0];                 // lane l holds colsum[l & 15]
  s += __shfl_xor(s, 1, 32);
  s += __shfl_xor(s, 2, 32);
  s += __shfl_xor(s, 4, 32);
  s += __shfl_xor(s, 8, 32);
  return s;
#else
  float s = x;
#pragma unroll
  for (int off = 16; off > 0; off >>= 1) s += __shfl_xor(s, off, 32);
  return s;
#endif
}

__device__ __forceinline__ float l1d(float qx, float qy, float qz,
                                     float px, float py, float pz) {
  return fabsf(qx - px) + fabsf(qy - py) + fabsf(qz - pz);
}

struct Top8 { float d[K_NN]; int i[K_NN]; };

__device__ __forceinline__ void top8_update(Top8& t, float dist, int id) {
  if (dist < t.d[K_NN - 1]) {          // rarely taken after warm-up
    t.d[K_NN - 1] = dist; t.i[K_NN - 1] = id;
#pragma unroll
    for (int m = K_NN - 1; m > 0; --m) {
      if (t.d[m] < t.d[m - 1]) {
        float td = t.d[m]; t.d[m] = t.d[m - 1]; t.d[m - 1] = td;
        int   ti = t.i[m]; t.i[m] = t.i[m - 1]; t.i[m - 1] = ti;
      }
    }
  }
}

// ---------------- Kernel 1: K=8 self-KNN under L1 (pc1->nn1, pc2->nn2) -----
__global__ __launch_bounds__(BLK) void knn8_kernel(const float* __restrict__ pts,
                                                   int* __restrict__ nn) {
  __shared__ __align__(16) float tile[2][TILE_F];
  const int tid = threadIdx.x;
  const int i = blockIdx.x * BLK + tid;
  const float qx = pts[3 * i + 0];
  const float qy = pts[3 * i + 1];
  const float qz = pts[3 * i + 2];

  Top8 t;
#pragma unroll
  for (int k = 0; k < K_NN; ++k) { t.d[k] = 3.0e38f; t.i[k] = 0; }

  constexpr int NT = N_PTS / TILE;   // 8 tiles
  tile_fetch<VPT_KNN>(tile[0], pts, tid);
  int cur = 0;
  for (int tt = 0; tt < NT; ++tt) {
    if (tt + 1 < NT) {
      tile_fetch<VPT_KNN>(tile[cur ^ 1], pts + (tt + 1) * TILE_F, tid);
      ASYNC_WAIT(VPT_KNN);           // drain current tile, leave next in flight
    } else {
      ASYNC_WAIT(0);
    }
    __syncthreads();
    const float4* tl4 = (const float4*)tile[cur];
    const int base = tt * TILE;
    // 4 points = 12 floats = 3 x ds_load_b128, broadcast to all 32 lanes
    for (int j = 0; j < TILE; j += 4) {
      const float4 a = tl4[3 * (j >> 2) + 0];
      const float4 b = tl4[3 * (j >> 2) + 1];
      const float4 c = tl4[3 * (j >> 2) + 2];
      top8_update(t, l1d(qx, qy, qz, a.x, a.y, a.z), base + j + 0);
      top8_update(t, l1d(qx, qy, qz, a.w, b.x, b.y), base + j + 1);
      top8_update(t, l1d(qx, qy, qz, b.z, b.w, c.x), base + j + 2);
      top8_update(t, l1d(qx, qy, qz, c.y, c.z, c.w), base + j + 3);
    }
    __syncthreads();
    cur ^= 1;
  }
#pragma unroll
  for (int k = 0; k < K_NN; ++k) nn[i * K_NN + k] = t.i[k];
}

// ---------------- Kernel 2: argmin_{j} ||p1+flow - p2[j]||_1  -> idx0 ------
__global__ __launch_bounds__(BLK) void argmin_kernel(const float* __restrict__ p1,
                                                     const float* __restrict__ flow,
                                                     const float* __restrict__ p2,
                                                     int* __restrict__ idx0) {
  __shared__ __align__(16) float tile[2][TILE_F];
  const int tid = threadIdx.x;
  const int i = blockIdx.x * BLK + tid;
  const float qx = p1[3 * i + 0] + flow[3 * i + 0];
  const float qy = p1[3 * i + 1] + flow[3 * i + 1];
  const float qz = p1[3 * i + 2] + flow[3 * i + 2];

  float bestd = 3.0e38f; int besti = 0;
  constexpr int NT = N_PTS / TILE;
  tile_fetch<VPT_KNN>(tile[0], p2, tid);
  int cur = 0;
  for (int tt = 0; tt < NT; ++tt) {
    if (tt + 1 < NT) {
      tile_fetch<VPT_KNN>(tile[cur ^ 1], p2 + (tt + 1) * TILE_F, tid);
      ASYNC_WAIT(VPT_KNN);
    } else {
      ASYNC_WAIT(0);
    }
    __syncthreads();
    const float4* tl4 = (const float4*)tile[cur];
    const int base = tt * TILE;
    for (int j = 0; j < TILE; j += 4) {
      const float4 a = tl4[3 * (j >> 2) + 0];
      const float4 b = tl4[3 * (j >> 2) + 1];
      const float4 c = tl4[3 * (j >> 2) + 2];
      float d0 = l1d(qx, qy, qz, a.x, a.y, a.z);
      float d1 = l1d(qx, qy, qz, a.w, b.x, b.y);
      float d2 = l1d(qx, qy, qz, b.z, b.w, c.x);
      float d3 = l1d(qx, qy, qz, c.y, c.z, c.w);
      if (d0 < bestd) { bestd = d0; besti = base + j + 0; }
      if (d1 < bestd) { bestd = d1; besti = base + j + 1; }
      if (d2 < bestd) { bestd = d2; besti = base + j + 2; }
      if (d3 < bestd) { bestd = d3; besti = base + j + 3; }
    }
    __syncthreads();
    cur ^= 1;
  }
  idx0[i] = besti;
}

// ---------------- Kernel 3: deterministic dense segment-mean ----------------
// vec[j] = mean of flow[i] over {i : idx0[i]==j}, else 0.  (No float atomics.)
__global__ __launch_bounds__(BLK) void segmean_kernel(const int* __restrict__ idx0,
                                                      const float* __restrict__ flow,
                                                      float* __restrict__ vec) {
  __shared__ __align__(16) float tfl[2][TSEG * 3];
  __shared__ __align__(16) int   tix[2][TSEG];
  const int tid = threadIdx.x;
  const int j = blockIdx.x * BLK + tid;

  float cnt = 0.f, sx = 0.f, sy = 0.f, sz = 0.f;
  constexpr int NT = N_PTS / TSEG;   // 16 tiles
  tile_fetch<VF_SEG>(tfl[0], flow, tid);
  tile_fetch<VI_SEG>((float*)tix[0], (const float*)idx0, tid);
  int cur = 0;
  for (int t = 0; t < NT; ++t) {
    if (t + 1 < NT) {
      tile_fetch<VF_SEG>(tfl[cur ^ 1], flow + (t + 1) * TSEG * 3, tid);
      tile_fetch<VI_SEG>((float*)tix[cur ^ 1], (const float*)(idx0 + (t + 1) * TSEG), tid);
      ASYNC_WAIT(VF_SEG + VI_SEG);
    } else {
      ASYNC_WAIT(0);
    }
    __syncthreads();
    const float4* fl4 = (const float4*)tfl[cur];
    const int4*   ix4 = (const int4*)tix[cur];
    for (int s = 0; s < TSEG; s += 4) {
      const int4   ii = ix4[s >> 2];
      const float4 a  = fl4[3 * (s >> 2) + 0];
      const float4 b  = fl4[3 * (s >> 2) + 1];
      const float4 c  = fl4[3 * (s >> 2) + 2];
      float m;
      m = (ii.x == j) ? 1.0f : 0.0f;
      cnt += m; sx = fmaf(m, a.x, sx); sy = fmaf(m, a.y, sy); sz = fmaf(m, a.z, sz);
      m = (ii.y == j) ? 1.0f : 0.0f;
      cnt += m; sx = fmaf(m, a.w, sx); sy = fmaf(m, b.x, sy); sz = fmaf(m, b.y, sz);
      m = (ii.z == j) ? 1.0f : 0.0f;
      cnt += m; sx = fmaf(m, b.z, sx); sy = fmaf(m, b.w, sy); sz = fmaf(m, c.x, sz);
      m = (ii.w == j) ? 1.0f : 0.0f;
      cnt += m; sx = fmaf(m, c.y, sx); sy = fmaf(m, c.z, sy); sz = fmaf(m, c.w, sz);
    }
    __syncthreads();
    cur ^= 1;
  }
  const float inv = 1.0f / fmaxf(cnt, 1.0f);
  vec[3 * j + 0] = (cnt > 0.f) ? sx * inv : 0.0f;
  vec[3 * j + 1] = (cnt > 0.f) ? sy * inv : 0.0f;
  vec[3 * j + 2] = (cnt > 0.f) ? sz * inv : 0.0f;
}

// ---------------- Kernel 4: gather all loss terms, per-block partials -------
__global__ __launch_bounds__(BLK) void loss_kernel(const float* __restrict__ p1,
                                                   const float* __restrict__ p2,
                                                   const float* __restrict__ flow,
                                                   const int* __restrict__ nn1,
                                                   const int* __restrict__ nn2,
                                                   const int* __restrict__ idx0,
                                                   const float* __restrict__ vec,
                                                   float* __restrict__ part) {
  const int tid = threadIdx.x;
  const int i = blockIdx.x * BLK + tid;
  const float fx = flow[3 * i + 0], fy = flow[3 * i + 1], fz = flow[3 * i + 2];

  // smooth: neighbors 1..7 of nn1 (column 0 is self at distance 0)
  float sm = 0.f;
#pragma unroll
  for (int k = 1; k < K_NN; ++k) {
    const int g = nn1[K_NN * i + k];
    sm += l1d(fx, fy, fz, flow[3 * g + 0], flow[3 * g + 1], flow[3 * g + 2]);
  }

  const int j0 = idx0[i];
  // chamfer
  const float qx = p1[3 * i + 0] + fx, qy = p1[3 * i + 1] + fy, qz = p1[3 * i + 2] + fz;
  const float ch = l1d(qx, qy, qz, p2[3 * j0 + 0], p2[3 * j0 + 1], p2[3 * j0 + 2]);
  // flow-field consistency
  const float vx = vec[3 * j0 + 0], vy = vec[3 * j0 + 1], vz = vec[3 * j0 + 2];
  const float ffv = (vx - fx) * (vx - fx) + (vy - fy) * (vy - fy) + (vz - fz) * (vz - fz);
  // pc2 occupancy-masked term over nn2[idx0]
  float pnum = 0.f, pcnt = 0.f;
#pragma unroll
  for (int k = 0; k < K_NN; ++k) {
    const int g = nn2[K_NN * j0 + k];
    const float wx = vec[3 * g + 0], wy = vec[3 * g + 1], wz = vec[3 * g + 2];
    if (wx != 0.f && wy != 0.f && wz != 0.f) {
      const float dx = fx - wx, dy = fy - wy, dz = fz - wz;
      pnum += dx * dx + dy * dy + dz * dz;
      pcnt += 1.f;
    }
  }

  // block reduction: wave shuffles then 8-wave LDS combine
  float v0 = sm, v1 = ch, v2 = ffv, v3 = pnum, v4 = pcnt;
#pragma unroll
  for (int off = 16; off > 0; off >>= 1) {
    v0 += __shfl_xor(v0, off, 32);
    v1 += __shfl_xor(v1, off, 32);
    v2 += __shfl_xor(v2, off, 32);
    v3 += __shfl_xor(v3, off, 32);
    v4 += __shfl_xor(v4, off, 32);
  }
  __shared__ float red[5][BLK / 32];
  const int lane = tid & 31, w = tid >> 5;
  if (lane == 0) {
    red[0][w] = v0; red[1][w] = v1; red[2][w] = v2; red[3][w] = v3; red[4][w] = v4;
  }
  __syncthreads();
  if (tid < 5) {
    float s = 0.f;
#pragma unroll
    for (int ww = 0; ww < BLK / 32; ++ww) s += red[tid][ww];
    part[tid * NBLK + blockIdx.x] = s;
  }
}

// ---------------- Kernel 5: final WMMA-assisted reduction to scalar ---------
__global__ __launch_bounds__(32) void final_kernel(const float* __restrict__ part,
                                                   float* __restrict__ out) {
  const int lane = threadIdx.x;   // one full wave32, EXEC all ones for WMMA
  float c0 = part[0 * NBLK + lane] + part[0 * NBLK + lane + 32];
  float c1 = part[1 * NBLK + lane] + part[1 * NBLK + lane + 32];
  float c2 = part[2 * NBLK + lane] + part[2 * NBLK + lane + 32];
  float c3 = part[3 * NBLK + lane] + part[3 * NBLK + lane + 32];
  float c4 = part[4 * NBLK + lane] + part[4 * NBLK + lane + 32];
  c0 = wave_sum32(c0);
  c1 = wave_sum32(c1);
  c2 = wave_sum32(c2);
  c3 = wave_sum32(c3);
  c4 = wave_sum32(c4);
  if (lane == 0) {
    const float n = (float)N_PTS;
    const float total = c0 / (7.0f * n)          // smooth mean over N*7
                      + c1 / n                   // chamfer mean
                      + c2 / (3.0f * n)          // ff mean over N*3
                      + c3 / fmaxf(3.0f * c4, 1.0f); // pc2 masked mean
    out[0] = total;
  }
}

// ---------------- host-side launcher ----------------
extern "C" void kernel_launch(void* const* d_in, const int* in_sizes, int n_in,
                              void* d_out, int out_size, void* d_ws, size_t ws_size,
                              hipStream_t stream) {
  (void)in_sizes; (void)n_in; (void)out_size; (void)ws_size;
  const float* pc1  = (const float*)d_in[0];
  const float* pc2  = (const float*)d_in[1];
  const float* flow = (const float*)d_in[2];
  float* out = (float*)d_out;

  char* ws = (char*)d_ws;
  int*   nn1  = (int*)ws;   ws += (size_t)N_PTS * K_NN * sizeof(int);
  int*   nn2  = (int*)ws;   ws += (size_t)N_PTS * K_NN * sizeof(int);
  int*   idx0 = (int*)ws;   ws += (size_t)N_PTS * sizeof(int);
  float* vec  = (float*)ws; ws += (size_t)N_PTS * 3 * sizeof(float);
  float* part = (float*)ws; // 5 * NBLK floats

  dim3 grid(NBLK), block(BLK);
  knn8_kernel  <<<grid, block, 0, stream>>>(pc1, nn1);
  knn8_kernel  <<<grid, block, 0, stream>>>(pc2, nn2);
  argmin_kernel<<<grid, block, 0, stream>>>(pc1, flow, pc2, idx0);
  segmean_kernel<<<grid, block, 0, stream>>>(idx0, flow, vec);
  loss_kernel  <<<grid, block, 0, stream>>>(pc1, pc2, flow, nn1, nn2, idx0, vec, part);
  final_kernel <<<dim3(1), dim3(32), 0, stream>>>(part, out);
}